// Head_163208757259
// MI455X (gfx1250) — compile-verified
//
#include <hip/hip_runtime.h>

typedef __attribute__((ext_vector_type(2))) float v2f;
typedef __attribute__((ext_vector_type(8))) float v8f;

static constexpr int Bsz = 4;
static constexpr int Tsz = 4096;
static constexpr int Csz = 1024;
static constexpr int Hsz = 64;
static constexpr int BT  = Bsz * Tsz;   // 16384 rows

// ---------------------------------------------------------------------------
// Kernel 1: q/k/v projections.  One wave computes a 16-row tile of x against
// all three weight matrices using V_WMMA_F32_16X16X4_F32 (fp32 matrix pipe).
// A-fragment layout (ISA 7.12.2, 32-bit A 16x4): lane m=lane&15 holds row m,
// K = 2*(lane>>4)+{0,1}  -> one contiguous float2 load per K-step.
// ---------------------------------------------------------------------------
__global__ __launch_bounds__(128) void proj_qkv(
    const float* __restrict__ x,
    const float* __restrict__ Wk,
    const float* __restrict__ Wq,
    const float* __restrict__ Wv,
    float* __restrict__ qo, float* __restrict__ ko, float* __restrict__ vo)
{
  const int lane = threadIdx.x & 31;
  const int wid  = threadIdx.x >> 5;
  const int tile = blockIdx.x * 4 + wid;   // 16-row tile index
  const int row0 = tile * 16;
  const int m    = lane & 15;
  const int kp   = lane >> 4;              // K-pair selector (0/1)

  v8f qa[4] = {}, ka[4] = {}, va[4] = {};

  const float* xr = x + (size_t)(row0 + m) * Csz + 2 * kp;

  for (int kk = 0; kk < Csz; kk += 4) {
    v2f a = *(const v2f*)(xr + kk);
    const int kr = kk + 2 * kp;
#pragma unroll
    for (int nn = 0; nn < 4; ++nn) {
      const int n = nn * 16 + m;
      v2f b;
      b.x = Wq[(size_t)kr * Hsz + n];
      b.y = Wq[(size_t)(kr + 1) * Hsz + n];
      qa[nn] = __builtin_amdgcn_wmma_f32_16x16x4_f32(false, a, false, b,
                                                     (short)0, qa[nn], false, false);
      b.x = Wk[(size_t)kr * Hsz + n];
      b.y = Wk[(size_t)(kr + 1) * Hsz + n];
      ka[nn] = __builtin_amdgcn_wmma_f32_16x16x4_f32(false, a, false, b,
                                                     (short)0, ka[nn], false, false);
      b.x = Wv[(size_t)kr * Hsz + n];
      b.y = Wv[(size_t)(kr + 1) * Hsz + n];
      va[nn] = __builtin_amdgcn_wmma_f32_16x16x4_f32(false, a, false, b,
                                                     (short)0, va[nn], false, false);
    }
  }

  // C/D layout: VGPR r, lanes 0-15 -> row r, lanes 16-31 -> row r+8.
#pragma unroll
  for (int nn = 0; nn < 4; ++nn)
#pragma unroll
    for (int r = 0; r < 8; ++r) {
      const size_t idx = (size_t)(row0 + r + 8 * kp) * Hsz + nn * 16 + m;
      qo[idx] = qa[nn][r];
      ko[idx] = ka[nn][r];
      vo[idx] = va[nn][r];
    }
}

// ---------------------------------------------------------------------------
// Kernel 2: causal flash-attention forward.  One wave owns one 16-row q tile,
// streams causal k/v tiles (L2-resident: k+v = 8 MB << 192 MB L2).
// S = Q K^T and O += P V both on V_WMMA_F32_16X16X4_F32.
// Online softmax stats kept per-row in registers; P converted from the WMMA
// C-layout to the A-layout via a per-wave padded LDS tile (DS ops are
// in-order within a wave, so only a compiler wave_barrier is required —
// a block barrier would deadlock since causal trip counts differ per wave).
// ---------------------------------------------------------------------------
__global__ __launch_bounds__(128) void flash_attn(
    const float* __restrict__ qm, const float* __restrict__ km,
    const float* __restrict__ vm, float* __restrict__ out)
{
  __shared__ float pbuf[4][16 * 17];       // per-wave P bounce tile, padded

  const int lane = threadIdx.x & 31;
  const int wid  = threadIdx.x >> 5;
  const int tile = blockIdx.x * 4 + wid;   // global 16-row q tile
  const int bb   = tile >> 8;              // batch (256 tiles per batch)
  const int ti   = tile & 255;             // q-tile index within batch
  const int m    = lane & 15;
  const int kp   = lane >> 4;

  const size_t base = (size_t)bb * Tsz * Hsz;
  const float* qb = qm + base;
  const float* kb = km + base;
  const float* vb = vm + base;

  // Q tile as A-fragments for all 16 K-steps, softmax scale folded in once.
  v2f qf[16];
  {
    const float* qr = qb + (size_t)(ti * 16 + m) * Hsz + 2 * kp;
#pragma unroll
    for (int s = 0; s < 16; ++s) {
      v2f t = *(const v2f*)(qr + 4 * s);
      qf[s].x = t.x * 0.125f;              // 1/sqrt(64)
      qf[s].y = t.y * 0.125f;
    }
  }

  v8f acc[4] = {};
  float mr[8], lr[8];
#pragma unroll
  for (int r = 0; r < 8; ++r) { mr[r] = -1e30f; lr[r] = 0.0f; }

  float* pb = pbuf[wid];

  for (int j = 0; j <= ti; ++j) {
    // ---- S = (Q*scale) . K^T : B-fragment is contiguous float2 from K rows
    v8f s8 = {};
    const float* kt = kb + (size_t)(j * 16 + m) * Hsz + 2 * kp;
#pragma unroll
    for (int s = 0; s < 16; ++s) {
      v2f bf = *(const v2f*)(kt + 4 * s);
      s8 = __builtin_amdgcn_wmma_f32_16x16x4_f32(false, qf[s], false, bf,
                                                 (short)0, s8, false, false);
    }

    // ---- causal mask on diagonal tile (C layout: row = r+8*kp, col = m)
    if (j == ti) {
#pragma unroll
      for (int r = 0; r < 8; ++r)
        if (m > r + 8 * kp) s8[r] = -1e30f;
    }

    __builtin_amdgcn_wave_barrier();       // WAR on pb vs previous iteration

    // ---- online softmax (reductions within 16-lane halves: wave32)
#pragma unroll
    for (int r = 0; r < 8; ++r) {
      float rm = s8[r];
#pragma unroll
      for (int off = 1; off < 16; off <<= 1)
        rm = fmaxf(rm, __shfl_xor(rm, off, 32));
      const float mnew  = fmaxf(mr[r], rm);
      const float alpha = __expf(mr[r] - mnew);
      mr[r] = mnew;
      const float p = __expf(s8[r] - mnew);
      float rs = p;
#pragma unroll
      for (int off = 1; off < 16; off <<= 1)
        rs += __shfl_xor(rs, off, 32);
      lr[r] = lr[r] * alpha + rs;
#pragma unroll
      for (int nn = 0; nn < 4; ++nn) acc[nn][r] *= alpha;
      pb[(r + 8 * kp) * 17 + m] = p;       // C-layout -> LDS
    }

    __builtin_amdgcn_wave_barrier();       // RAW: P stores before A-frag loads

    // ---- P as A-fragments (4 K-steps of 4)
    v2f pf[4];
#pragma unroll
    for (int s = 0; s < 4; ++s) {
      pf[s].x = pb[m * 17 + 4 * s + 2 * kp];
      pf[s].y = pb[m * 17 + 4 * s + 2 * kp + 1];
    }

    // ---- O += P . V
    const float* vt = vb + (size_t)(j * 16) * Hsz;
#pragma unroll
    for (int s = 0; s < 4; ++s) {
      const int kr = 4 * s + 2 * kp;
#pragma unroll
      for (int nn = 0; nn < 4; ++nn) {
        const int n = nn * 16 + m;
        v2f bf;
        bf.x = vt[(size_t)kr * Hsz + n];
        bf.y = vt[(size_t)(kr + 1) * Hsz + n];
        acc[nn] = __builtin_amdgcn_wmma_f32_16x16x4_f32(false, pf[s], false, bf,
                                                        (short)0, acc[nn], false, false);
      }
    }
  }

  // ---- epilogue: normalize and store
#pragma unroll
  for (int nn = 0; nn < 4; ++nn)
#pragma unroll
    for (int r = 0; r < 8; ++r) {
      const size_t idx = base + (size_t)(ti * 16 + r + 8 * kp) * Hsz + nn * 16 + m;
      out[idx] = acc[nn][r] / lr[r];
    }
}

// ---------------------------------------------------------------------------
extern "C" void kernel_launch(void* const* d_in, const int* in_sizes, int n_in,
                              void* d_out, int out_size, void* d_ws, size_t ws_size,
                              hipStream_t stream) {
  (void)in_sizes; (void)n_in; (void)out_size; (void)ws_size;

  const float* x  = (const float*)d_in[0];
  const float* Wk = (const float*)d_in[1];
  const float* Wq = (const float*)d_in[2];
  const float* Wv = (const float*)d_in[3];
  float* out = (float*)d_out;

  float* qo = (float*)d_ws;                       // 4 MB
  float* ko = qo + (size_t)BT * Hsz;              // 4 MB
  float* vo = ko + (size_t)BT * Hsz;              // 4 MB (12 MB total scratch)

  const int tiles = BT / 16;                      // 1024 wave-tiles
  dim3 block(128);                                // 4 waves (wave32)
  dim3 grid(tiles / 4);                           // 256 blocks

  proj_qkv<<<grid, block, 0, stream>>>(x, Wk, Wq, Wv, qo, ko, vo);
  flash_attn<<<grid, block, 0, stream>>>(qo, ko, vo, out);
}